// SequentialParsingModel_72224170049756
// MI455X (gfx1250) — compile-verified
//
#include <hip/hip_runtime.h>
#include <hip/hip_bf16.h>

// ---- problem constants (from reference) ----
#define BB   32
#define SS   256
#define NT   (BB*SS)      // 8192 tokens
#define DD   300
#define PDIM 32
#define MM   66
#define FF   398          // D + PD + M
#define FPAD 400          // padded K for WMMA (K multiple of 4)
#define HH   6
#define G4   24           // 4*H
#define AA   12

typedef __attribute__((ext_vector_type(2))) float v2f;
typedef __attribute__((ext_vector_type(8))) float v8f;

__device__ __forceinline__ float fast_exp(float x)  { return __expf(x); }
__device__ __forceinline__ float fast_rcp(float x)  { return __builtin_amdgcn_rcpf(x); }
__device__ __forceinline__ float fast_tanh(float x) {
    // tanh(x) = 1 - 2/(e^{2x}+1); robust at +/-inf of the exp
    float e = fast_exp(2.0f * x);
    return 1.0f - 2.0f * fast_rcp(e + 1.0f);
}
__device__ __forceinline__ float fast_sig(float x) {
    return fast_rcp(1.0f + fast_exp(-x));
}
__device__ __forceinline__ float wave_max(float v) {
    for (int o = 16; o > 0; o >>= 1) v = fmaxf(v, __shfl_xor(v, o, 32));
    return v;
}
__device__ __forceinline__ float wave_sum(float v) {
    for (int o = 16; o > 0; o >>= 1) v += __shfl_xor(v, o, 32);
    return v;
}

// ---------------------------------------------------------------------------
// K1: embedding gather + concat into padded feats [NT][FPAD]; pack Wx_f|Wx_b
//     into padded wxall [FPAD][48]; zero the loss accumulator.
// ---------------------------------------------------------------------------
__global__ void k_prep(const int* __restrict__ words, const int* __restrict__ pos,
                       const float* __restrict__ morph,
                       const float* __restrict__ word_table, const float* __restrict__ pos_table,
                       const float* __restrict__ Wx_f, const float* __restrict__ Wx_b,
                       float* __restrict__ feats, float* __restrict__ wxall,
                       float* __restrict__ loss_out) {
    long e = (long)blockIdx.x * blockDim.x + threadIdx.x;
    if (e == 0) loss_out[0] = 0.0f;
    const long NF = (long)NT * FPAD;
    if (e < NF) {
        int t = (int)(e / FPAD), c = (int)(e % FPAD);
        float val;
        if (c < DD)              val = word_table[(long)words[t] * DD + c];
        else if (c < DD + PDIM)  val = pos_table[pos[t] * PDIM + (c - DD)];
        else if (c < FF)         val = morph[(long)t * MM + (c - DD - PDIM)];
        else                     val = 0.0f;
        feats[e] = val;
    } else {
        long e2 = e - NF;
        if (e2 < (long)FPAD * 48) {
            int r = (int)(e2 / 48), c = (int)(e2 % 48);
            float val = 0.0f;
            if (r < FF) val = (c < G4) ? Wx_f[r * G4 + c] : Wx_b[r * G4 + (c - G4)];
            wxall[e2] = val;
        }
    }
}

// ---------------------------------------------------------------------------
// K2: xg[NT][48] = feats[NT][FPAD] @ wxall[FPAD][48] + bias, via
//     V_WMMA_F32_16X16X4_F32.  One wave per 16x16 tile; 512 M-tiles x 3 N-tiles.
//     A 16x4: lanes 0-15 hold K=k,k+1 ; lanes 16-31 hold K=k+2,k+3.
//     B 4x16: same K split, column = lane&15.  D: vgpr r -> M = (lane>=16?8:0)+r.
// ---------------------------------------------------------------------------
__global__ void k_gemm_wmma(const float* __restrict__ feats, const float* __restrict__ wxall,
                            const float* __restrict__ b_f, const float* __restrict__ b_b,
                            float* __restrict__ xg) {
    int wid  = (int)((blockIdx.x * blockDim.x + threadIdx.x) >> 5);
    int lane = threadIdx.x & 31;
    int nt = wid % 3, mt = wid / 3;           // mt in [0,512)
    int half = lane >> 4, lm = lane & 15;
    int row = mt * 16 + lm;
    int col = nt * 16 + lm;                   // < 48 always
    const float* arow = feats + (long)row * FPAD;
    v8f acc = {};
    for (int k = 0; k < FPAD; k += 4) {
        int kk = k + 2 * half;
        v2f a, b;
        a.x = arow[kk];
        a.y = arow[kk + 1];
        b.x = wxall[kk * 48 + col];
        b.y = wxall[(kk + 1) * 48 + col];
        acc = __builtin_amdgcn_wmma_f32_16x16x4_f32(false, a, false, b,
                                                    (short)0, acc, false, false);
    }
    float bias = (col < G4) ? b_f[col] : b_b[col - G4];
    #pragma unroll
    for (int r = 0; r < 8; ++r) {
        int orow = mt * 16 + half * 8 + r;
        xg[(long)orow * 48 + col] = acc[r] + bias;
    }
}

// ---------------------------------------------------------------------------
// K3: sequential LSTM scans, one wave per (direction, batch sample) = 64
//     independent waves, NO barriers, NO LDS.  h is replicated across lanes
//     in registers; each of lanes 0..23 owns one gate column of Wh; the
//     gate gather and h broadcast use intra-wave shuffles.  The xg load for
//     step t+1 is software-pipelined (independent of the recurrence).
//     Gate order (keras): i, f, cand, o.  hs[(dir*NT + b*S + t)*6 + j].
// ---------------------------------------------------------------------------
__global__ void k_lstm(const float* __restrict__ xg,
                       const float* __restrict__ Wh_f, const float* __restrict__ Wh_b,
                       float* __restrict__ hs) {
    int wid  = (int)((blockIdx.x * blockDim.x + threadIdx.x) >> 5); // 0..63
    int lane = threadIdx.x & 31;
    int dir  = wid >> 5;      // 0 = fwd, 1 = bwd
    int b_   = wid & 31;
    const float* Wh = dir ? Wh_b : Wh_f;
    int u = (lane < G4) ? lane : 0;           // gate column owned by this lane
    float wh[HH];
    #pragma unroll
    for (int k = 0; k < HH; ++k) wh[k] = Wh[k * G4 + u];
    float h[HH];
    #pragma unroll
    for (int k = 0; k < HH; ++k) h[k] = 0.0f;
    float c = 0.0f;                           // valid on lanes 0..5 only

    const float* xbase = xg + (long)b_ * SS * 48 + dir * G4 + u;
    int t0 = dir ? (SS - 1) : 0;
    float gnext = xbase[(long)t0 * 48];
    for (int step = 0; step < SS; ++step) {
        int t = dir ? (SS - 1 - step) : step;
        float g = gnext;
        if (step + 1 < SS) {
            int tn = dir ? (SS - 2 - step) : (step + 1);
            gnext = xbase[(long)tn * 48];     // independent of recurrence
        }
        #pragma unroll
        for (int k = 0; k < HH; ++k) g = fmaf(h[k], wh[k], g);
        // lanes 0..11 and 18..23 hold sigmoid gates, 12..17 hold tanh(cand)
        float a = (lane < 2 * HH || lane >= 3 * HH) ? fast_sig(g) : fast_tanh(g);
        float f    = __shfl(a, lane + HH,     32);
        float cand = __shfl(a, lane + 2 * HH, 32);
        float o    = __shfl(a, lane + 3 * HH, 32);
        float hn = 0.0f;
        if (lane < HH) {
            c  = f * c + a * cand;            // a == sigmoid(i) on lanes 0..5
            hn = o * fast_tanh(c);
            hs[((long)dir * NT + (long)b_ * SS + t) * HH + lane] = hn;
        }
        #pragma unroll
        for (int k = 0; k < HH; ++k) h[k] = __shfl(hn, k, 32);
    }
}

// ---------------------------------------------------------------------------
// K4: ua/wa = concat(hf,hb) @ W(12x12) + bias, one thread per token.
// ---------------------------------------------------------------------------
__global__ void k_uawa(const float* __restrict__ hs,
                       const float* __restrict__ Wu, const float* __restrict__ bu,
                       const float* __restrict__ Ww, const float* __restrict__ bw,
                       float* __restrict__ ua, float* __restrict__ wa) {
    int t = blockIdx.x * blockDim.x + threadIdx.x;
    if (t >= NT) return;
    float r[AA];
    #pragma unroll
    for (int k = 0; k < HH; ++k) {
        r[k]      = hs[(long)t * HH + k];
        r[HH + k] = hs[((long)NT + t) * HH + k];
    }
    #pragma unroll
    for (int d = 0; d < AA; ++d) {
        float su = bu[d], sw = bw[d];
        #pragma unroll
        for (int k = 0; k < AA; ++k) {
            su += r[k] * Wu[k * AA + d];
            sw += r[k] * Ww[k * AA + d];
        }
        ua[(long)t * AA + d] = su;
        wa[(long)t * AA + d] = sw;
    }
}

// ---------------------------------------------------------------------------
// K5: one wave per (b,i) row. scores[j] = v . tanh(ua[b,j] + wa[b,i]),
//     diagonal masked to -10000; fused log-softmax + CE (atomic loss) +
//     exp(scores) table write: d_out[1 + ((i-1)*B + b)*S + j].
// ---------------------------------------------------------------------------
__global__ void k_scores(const float* __restrict__ ua, const float* __restrict__ wa,
                         const float* __restrict__ vvec, const int* __restrict__ heads,
                         float* __restrict__ out) {
    int wid  = (int)((blockIdx.x * blockDim.x + threadIdx.x) >> 5); // = b*S + i
    int lane = threadIdx.x & 31;
    int b_ = wid >> 8, i = wid & 255;
    float wai[AA], v12[AA];
    const float* wp = wa + (long)wid * AA;
    #pragma unroll
    for (int d = 0; d < AA; ++d) { wai[d] = wp[d]; v12[d] = vvec[d]; }
    float sc8[8];
    float m = -1e30f;
    #pragma unroll
    for (int jj = 0; jj < 8; ++jj) {
        int j = jj * 32 + lane;
        const float* up = ua + ((long)b_ * SS + j) * AA;
        if (jj + 1 < 8)  // pull next column's ua row while this tanh chain runs
            __builtin_prefetch(ua + ((long)b_ * SS + j + 32) * AA, 0, 3);
        float s = 0.0f;
        #pragma unroll
        for (int d = 0; d < AA; ++d) s += fast_tanh(up[d] + wai[d]) * v12[d];
        if (j == i) s = -10000.0f;
        sc8[jj] = s;
        m = fmaxf(m, s);
    }
    m = wave_max(m);
    float se = 0.0f;
    #pragma unroll
    for (int jj = 0; jj < 8; ++jj) se += fast_exp(sc8[jj] - m);
    se = wave_sum(se);
    float logZ = m + __logf(se);
    int head = heads[wid];
    float shd = 0.0f;
    #pragma unroll
    for (int jj = 0; jj < 8; ++jj) if (jj * 32 + lane == head) shd = sc8[jj];
    shd = wave_sum(shd);
    if (i >= 1) {
        if (lane == 0) atomicAdd(out, (logZ - shd) * (1.0f / (float)BB));
        float* trow = out + 1 + ((long)(i - 1) * BB + b_) * SS;
        #pragma unroll
        for (int jj = 0; jj < 8; ++jj) trow[jj * 32 + lane] = fast_exp(sc8[jj]);
    }
}

extern "C" void kernel_launch(void* const* d_in, const int* in_sizes, int n_in,
                              void* d_out, int out_size, void* d_ws, size_t ws_size,
                              hipStream_t stream) {
    (void)in_sizes; (void)n_in; (void)out_size; (void)ws_size;
    const int*   words      = (const int*)  d_in[0];
    const int*   pos        = (const int*)  d_in[1];
    const float* morph      = (const float*)d_in[2];
    const int*   heads      = (const int*)  d_in[3];
    const float* word_table = (const float*)d_in[4];
    const float* pos_table  = (const float*)d_in[5];
    const float* Wx_f       = (const float*)d_in[6];
    const float* Wh_f       = (const float*)d_in[7];
    const float* b_f        = (const float*)d_in[8];
    const float* Wx_b       = (const float*)d_in[9];
    const float* Wh_b       = (const float*)d_in[10];
    const float* b_b        = (const float*)d_in[11];
    const float* Wu         = (const float*)d_in[12];
    const float* bu         = (const float*)d_in[13];
    const float* Ww         = (const float*)d_in[14];
    const float* bw         = (const float*)d_in[15];
    const float* vvec       = (const float*)d_in[16];
    float* out = (float*)d_out;

    // workspace layout (floats)
    float* ws    = (float*)d_ws;
    float* feats = ws;                                   // NT*FPAD
    float* wxall = feats + (long)NT * FPAD;              // FPAD*48
    float* xg    = wxall + (long)FPAD * 48;              // NT*48
    float* hs    = xg    + (long)NT * 48;                // 2*NT*6
    float* ua    = hs    + 2L * NT * HH;                 // NT*12
    float* wa    = ua    + (long)NT * AA;                // NT*12

    long n1 = (long)NT * FPAD + (long)FPAD * 48;
    k_prep<<<(int)((n1 + 255) / 256), 256, 0, stream>>>(
        words, pos, morph, word_table, pos_table, Wx_f, Wx_b, feats, wxall, out);

    // 512 M-tiles * 3 N-tiles = 1536 waves; 8 waves/block -> 192 blocks
    k_gemm_wmma<<<192, 256, 0, stream>>>(feats, wxall, b_f, b_b, xg);

    // 64 independent waves (2 dirs * 32 samples), 8 waves/block -> 8 blocks
    k_lstm<<<8, 256, 0, stream>>>(xg, Wh_f, Wh_b, hs);

    k_uawa<<<NT / 256, 256, 0, stream>>>(hs, Wu, bu, Ww, bw, ua, wa);

    // 8192 waves (one per (b,i)) -> 1024 blocks of 8 waves
    k_scores<<<1024, 256, 0, stream>>>(ua, wa, vvec, heads, out);
}